// CASSI_22230750724218
// MI455X (gfx1250) — compile-verified
//
#include <hip/hip_runtime.h>

// CASSI forward on MI455X (gfx1250). Memory-bound (AI ~0.5 FLOP/B) =>
// built around CDNA5 async global->LDS DMA (ASYNCcnt), a 4-slot circular
// LDS column window (power-of-two => no halo-pointer selects), scatter->
// gather inversion (no atomics), and output-PAIRED conflict-free LDS gathers
// (adjacent x words, one shared ca[n] per column).

#define M_ROWS   1024
#define N_COLS   1024
#define L_BANDS  64
#define N_OUT    (N_COLS + L_BANDS - 1)   // 1087
#define CHUNK    128                      // x columns staged per pipeline stage
#define NCHUNK   (N_COLS / CHUNK)         // 8
#define CHUNK_F  (CHUNK * L_BANDS)        // 8192 floats = 32 KB
#define XCOLS    512                      // 4-slot circular window (power of 2)
#define BLOCK    64                       // 2 waves; each thread owns 2 outputs
// LDS: circular x window (128 KB) + ca row (4 KB)
#define SMEM_FLOATS (XCOLS * L_BANDS + N_COLS)
#define SMEM_BYTES  (SMEM_FLOATS * 4)     // 135168 B < 160 KB CU-mode budget

// Builtin parameter types per hipcc diagnostic: pointer to 16-byte int vector,
// addrspace(1) global source, addrspace(3) LDS destination.
typedef int v4i __attribute__((vector_size(4 * sizeof(int))));
typedef __attribute__((address_space(1))) v4i gv4i_t;
typedef __attribute__((address_space(3))) v4i lv4i_t;
typedef __attribute__((address_space(3))) void lds_void_t;

// ---- CDNA5 async DMA: global -> LDS, 16 bytes per lane, tracked by ASYNCcnt
__device__ __forceinline__ void async_b128(const float* g, float* l) {
#if __has_builtin(__builtin_amdgcn_global_load_async_to_lds_b128)
  __builtin_amdgcn_global_load_async_to_lds_b128(
      (gv4i_t*)g, (lv4i_t*)l, /*offset=*/0, /*cpol=*/0);
#else
  asm volatile("global_load_async_to_lds_b128 %0, %1, off"
               :: "v"((lds_void_t*)l), "v"(g) : "memory");
#endif
}

__device__ __forceinline__ void wait_async_le32() {
#if __has_builtin(__builtin_amdgcn_s_wait_asynccnt)
  __builtin_amdgcn_s_wait_asynccnt(32);
#else
  asm volatile("s_wait_asynccnt 32" ::: "memory");
#endif
}

__device__ __forceinline__ void wait_async_le0() {
#if __has_builtin(__builtin_amdgcn_s_wait_asynccnt)
  __builtin_amdgcn_s_wait_asynccnt(0);
#else
  asm volatile("s_wait_asynccnt 0" ::: "memory");
#endif
}

// Issue one 32 KB x-chunk (CHUNK cols x 64 bands) into circular slot c&3.
// 2048 float4 transfers / 64 threads = 32 async b128 ops per thread.
__device__ __forceinline__ void issue_chunk(const float* __restrict__ xrow,
                                            float* __restrict__ xc,
                                            int c, int tid) {
  const float* g = xrow + c * CHUNK_F;
  float*       s = xc + (c & 3) * CHUNK_F;
#pragma unroll
  for (int i = 0; i < 32; ++i) {
    const int f4 = tid + i * BLOCK;        // float4 index within chunk
    async_b128(g + f4 * 4, s + f4 * 4);
  }
}

// Paired anti-diagonal gather for outputs (j, j+1), j even:
//   column n contributes x[n][j-n] -> out j and x[n][j+1-n] -> out j+1
//   (ADJACENT LDS words), both scaled by the same ca[n].
// Circular window: column n lives at ((n & 511) << 6) + l  (no halo select).
// Skewed order i=(tid+s)&63 => x banks (63-i,64-i) and ca bank (2tid+i)%64
// are distinct across all 32 lanes. Covers out j bands l=0..63 and out j+1
// bands l'=1..63 (the l'=0 edge is handled by the caller). When MASKED,
// out-of-range columns wrap inside LDS and only the FMAs are suppressed.
template <bool MASKED>
__device__ __forceinline__ void pair_sum(int j, int tid,
                                         const float* __restrict__ xc,
                                         const float* __restrict__ cas,
                                         float& out0, float& out1) {
  float acc0 = 0.0f, acc1 = 0.0f;
#pragma unroll
  for (int s = 0; s < L_BANDS; ++s) {
    const int i = (tid + s) & (L_BANDS - 1);
    const int n = j - (L_BANDS - 1) + i;   // column
    const int l = (L_BANDS - 1) - i;       // band feeding out j
    const bool ok = !MASKED || ((unsigned)n < (unsigned)N_COLS);
    const int idx = ((n & (XCOLS - 1)) << 6) + l;
    const float x0 = xc[idx];                          // -> out j   (band l)
    const float x1 = xc[idx + 1];                      // -> out j+1 (band l+1)
    const float c  = cas[MASKED ? (n & (N_COLS - 1)) : n];
    if (ok)                        acc0 += x0 * c;
    if (ok && l < (L_BANDS - 1))   acc1 += x1 * c;     // l==63 pair is invalid
  }
  out0 = acc0;
  out1 = acc1;
}

__global__ __launch_bounds__(BLOCK, 1)
void cassi_fwd_kernel(const float* __restrict__ x,
                      const float* __restrict__ ca,
                      float* __restrict__ y) {
  extern __shared__ float smem[];
  float* xc  = smem;                       // XCOLS * 64 floats (circular)
  float* cas = smem + XCOLS * L_BANDS;     // N_COLS floats

  const int m   = blockIdx.x;
  const int tid = threadIdx.x;

  const float* xrow  = x  + (size_t)m * (N_COLS * L_BANDS);
  const float* carow = ca + (size_t)m * N_COLS;
  float*       yrow  = y  + (size_t)m * N_OUT;

  // Prologue: chunk0 -> slot0, full ca row, chunk1 -> slot1 (all async DMA).
  issue_chunk(xrow, xc, 0, tid);
#pragma unroll
  for (int i = 0; i < 4; ++i) {
    const int f4 = tid + i * BLOCK;        // 256 float4 = 1024 ca floats
    async_b128(carow + f4 * 4, cas + f4 * 4);
  }
  issue_chunk(xrow, xc, 1, tid);

#pragma unroll 1
  for (int k = 0; k < NCHUNK; ++k) {
    // Async loads complete in order: <=32 outstanding means everything up to
    // and including chunk k (+ca) has landed; chunk k+1 may still be in flight.
    if (k < NCHUNK - 1) wait_async_le32(); else wait_async_le0();
    __syncthreads();

    // Each thread owns the output pair (j, j+1): 64 pairs cover the chunk.
    const int j = k * CHUNK + 2 * tid;
    float a0, a1;
    if (k == 0) pair_sum<true >(j, tid, xc, cas, a0, a1);
    else        pair_sum<false>(j, tid, xc, cas, a0, a1);
    // Edge term for out j+1 at band l'=0: column n=j+1 (always resident).
    a1 += xc[(((j + 1) & (XCOLS - 1)) << 6)] * cas[j + 1];
    __builtin_nontemporal_store(a0, yrow + j);
    __builtin_nontemporal_store(a1, yrow + j + 1);

    // Dispersion tail j in [N, N+63): contributing n in [961,1023] all live
    // in the final chunk's slot, so the same paired gather handles it.
    if (k == NCHUNK - 1 && tid < 32) {
      const int jt = N_COLS + 2 * tid;
      float b0, b1;
      pair_sum<true>(jt, tid, xc, cas, b0, b1);
      __builtin_nontemporal_store(b0, yrow + jt);
      if (jt + 1 < N_OUT) __builtin_nontemporal_store(b1, yrow + jt + 1);
    }

    __syncthreads();  // all reads of slot (k-2)&3 done before refilling it
    if (k + 2 < NCHUNK) issue_chunk(xrow, xc, k + 2, tid);
  }
}

extern "C" void kernel_launch(void* const* d_in, const int* in_sizes, int n_in,
                              void* d_out, int out_size, void* d_ws, size_t ws_size,
                              hipStream_t stream) {
  (void)in_sizes; (void)n_in; (void)out_size; (void)d_ws; (void)ws_size;
  const float* x  = (const float*)d_in[0];   // (1, 1024, 1024, 64) f32
  const float* ca = (const float*)d_in[1];   // (1, 1024, 1024, 1)  f32
  float*       y  = (float*)d_out;           // (1, 1024, 1087, 1)  f32

  dim3 grid(M_ROWS);
  dim3 block(BLOCK);
  hipLaunchKernelGGL(cassi_fwd_kernel, grid, block, SMEM_BYTES, stream, x, ca, y);
}